// Mean_nBatch_78950088835540
// MI455X (gfx1250) — compile-verified
//
#include <hip/hip_runtime.h>
#include <hip/hip_bf16.h>
#include <math.h>

typedef float v2f __attribute__((ext_vector_type(2)));
typedef float v8f __attribute__((ext_vector_type(8)));

#define SEGB 4096
#define FEAT 256
#define NOUT 4

// Workspace layout (bytes):
//   [0, 16K)          : int offsets[4096]
//   [16K, 17K)        : float zeropad[256]      (zeroed W rows for N >= NOUT)
//   [17K, 17K + 4M)   : float mean[4096 * 256]
#define WS_ZERO_OFF  (SEGB * sizeof(int))
#define WS_MEAN_OFF  (WS_ZERO_OFF + FEAT * sizeof(float))

// ---------------------------------------------------------------------------
// Phase 1: exclusive prefix sum of node_num (B=4096) with one 1024-thread WG.
// Also zero-fills the 256-float pad region used by the WMMA B-operand.
// ---------------------------------------------------------------------------
__global__ void Mean_nBatch_scan_kernel(const int* __restrict__ counts,
                                        int* __restrict__ offsets,
                                        float* __restrict__ zeropad) {
    __shared__ int sdata[1024];
    const int t = threadIdx.x;
    if (t < FEAT) zeropad[t] = 0.0f;

    int c0 = counts[t * 4 + 0];
    int c1 = counts[t * 4 + 1];
    int c2 = counts[t * 4 + 2];
    int c3 = counts[t * 4 + 3];
    sdata[t] = c0 + c1 + c2 + c3;
    __syncthreads();
    for (int off = 1; off < 1024; off <<= 1) {
        int v = sdata[t];
        int add = (t >= off) ? sdata[t - off] : 0;
        __syncthreads();
        sdata[t] = v + add;
        __syncthreads();
    }
    int excl = (t == 0) ? 0 : sdata[t - 1];
    offsets[t * 4 + 0] = excl;            excl += c0;
    offsets[t * 4 + 1] = excl;            excl += c1;
    offsets[t * 4 + 2] = excl;            excl += c2;
    offsets[t * 4 + 3] = excl;
}

// ---------------------------------------------------------------------------
// Phase 2: per-segment mean. One block per segment, 256 threads.
// Thread t: column chunk (t&63) as float4, row group (t>>6), stride-4 rows.
// Streaming coalesced b128 loads of x (the 268 MB phase).
// ---------------------------------------------------------------------------
__global__ void Mean_nBatch_segmean_kernel(const float* __restrict__ x,
                                           const int* __restrict__ counts,
                                           const int* __restrict__ offsets,
                                           float* __restrict__ mean) {
    __shared__ float4 red[256];
    const int b     = blockIdx.x;
    const int t     = threadIdx.x;
    const int chunk = t & 63;   // float4 chunk -> columns [4*chunk, 4*chunk+3]
    const int rgrp  = t >> 6;   // 0..3
    const int start = offsets[b];
    const int n     = counts[b];

    float4 acc = make_float4(0.f, 0.f, 0.f, 0.f);
    for (int r = rgrp; r < n; r += 4) {
        const float4 v =
            ((const float4*)(x + (size_t)(start + r) * FEAT))[chunk];
        acc.x += v.x; acc.y += v.y; acc.z += v.z; acc.w += v.w;
    }
    red[t] = acc;
    __syncthreads();

    if (rgrp == 0) {
        float4 s  = red[chunk];
        float4 s1 = red[chunk + 64];
        float4 s2 = red[chunk + 128];
        float4 s3 = red[chunk + 192];
        const float inv = 1.0f / (float)n;
        s.x = (s.x + s1.x + s2.x + s3.x) * inv;
        s.y = (s.y + s1.y + s2.y + s3.y) * inv;
        s.z = (s.z + s1.z + s2.z + s3.z) * inv;
        s.w = (s.w + s1.w + s2.w + s3.w) * inv;
        ((float4*)(mean + (size_t)b * FEAT))[chunk] = s;
    }
}

// ---------------------------------------------------------------------------
// Phase 3: projection sigmoid(mean @ W^T + b) with V_WMMA_F32_16X16X4_F32.
// One wave per tile of 16 segments. M = segment-in-tile, N = output (0..3
// live, 4..15 read a zero-filled pad row), K = feature dim in steps of 4.
// The zero pad makes every load unconditional: no EXEC churn in the loop,
// and EXEC stays all-1s as WMMA requires.
//
// ISA VGPR layouts (CDNA5 ISA):
//   A 16x4 f32: lane L, M = L%16; lanes 0-15 hold K=0,1; lanes 16-31 K=2,3.
//   B 4x16 f32: lane L, N = L%16; lanes 0-15 hold K=0,1; lanes 16-31 K=2,3.
//   C/D 16x16 f32: 8 VGPRs; element (M = v + 8*(L>=16), N = L%16).
// ---------------------------------------------------------------------------
__global__ void Mean_nBatch_proj_kernel(const float* __restrict__ mean,
                                        const float* __restrict__ W,
                                        const float* __restrict__ zeropad,
                                        const float* __restrict__ bias,
                                        float* __restrict__ out) {
    const int tile = blockIdx.x;          // 256 tiles of 16 segments
    const int lane = threadIdx.x;         // wave32
    const int mrow = lane & 15;           // M within tile (A-matrix row)
    const int half = lane >> 4;           // 0: K=0,1  1: K=2,3
    const int nco  = lane & 15;           // N for B and C/D

    const int seg = tile * 16 + mrow;
    const float2* __restrict__ arow = (const float2*)(mean + (size_t)seg * FEAT);
    const float2* __restrict__ wrow = (const float2*)(
        (nco < NOUT) ? (W + (size_t)nco * FEAT) : zeropad);

    v8f c = {};
    for (int k = 0; k < FEAT; k += 4) {
        const int pidx = (k >> 1) + half;       // float2 index -> elems k+2h, k+2h+1
        float2 aa = arow[pidx];
        float2 ww = wrow[pidx];
        v2f a;  a.x  = aa.x; a.y  = aa.y;
        v2f bm; bm.x = ww.x; bm.y = ww.y;
        // (neg_a, A, neg_b, B, c_mod, C, reuse_a, reuse_b)
        c = __builtin_amdgcn_wmma_f32_16x16x4_f32(
                false, a, false, bm, (short)0, c, false, false);
    }

    if (nco < NOUT) {
        const float bb = bias[nco];
#pragma unroll
        for (int v = 0; v < 8; ++v) {
            const int M = v + half * 8;
            const float val = c[v] + bb;
            out[(size_t)(tile * 16 + M) * NOUT + nco] =
                1.0f / (1.0f + __expf(-val));
        }
    }
}

// ---------------------------------------------------------------------------
extern "C" void kernel_launch(void* const* d_in, const int* in_sizes, int n_in,
                              void* d_out, int out_size, void* d_ws, size_t ws_size,
                              hipStream_t stream) {
    (void)in_sizes; (void)n_in; (void)out_size; (void)ws_size;

    const float* x        = (const float*)d_in[0];
    const int*   node_num = (const int*)d_in[1];
    const float* W        = (const float*)d_in[2];
    const float* bvec     = (const float*)d_in[3];
    float*       out      = (float*)d_out;

    int*   offsets = (int*)d_ws;
    float* zeropad = (float*)((char*)d_ws + WS_ZERO_OFF);
    float* mean    = (float*)((char*)d_ws + WS_MEAN_OFF);

    Mean_nBatch_scan_kernel<<<1, 1024, 0, stream>>>(node_num, offsets, zeropad);
    Mean_nBatch_segmean_kernel<<<SEGB, 256, 0, stream>>>(x, node_num, offsets, mean);
    Mean_nBatch_proj_kernel<<<SEGB / 16, 32, 0, stream>>>(mean, W, zeropad, bvec, out);
}